// GCN_755914244198
// MI455X (gfx1250) — compile-verified
//
#include <hip/hip_runtime.h>

#define NN 100000
#define NE 1600000
#define IN_D 128
#define HID_D 128
#define OUT_D 64

typedef float v2f __attribute__((ext_vector_type(2)));
typedef float v8f __attribute__((ext_vector_type(8)));

// ---------------- zero fill ----------------
__global__ void gcn_zero_f4(float4* __restrict__ p, int n4) {
  int i = blockIdx.x * blockDim.x + threadIdx.x;
  if (i < n4) p[i] = make_float4(0.f, 0.f, 0.f, 0.f);
}

// ---------------- degree count (int atomics) ----------------
__global__ void gcn_degrees(const int* __restrict__ src, const int* __restrict__ dst,
                            int* __restrict__ degO, int* __restrict__ degI, int ne) {
  int e = blockIdx.x * blockDim.x + threadIdx.x;
  if (e < ne) {
    atomicAdd(&degO[src[e]], 1);
    atomicAdd(&degI[dst[e]], 1);
  }
}

// ---------------- rsqrt(max(deg,1)) ----------------
__global__ void gcn_invsqrt(const int* __restrict__ degO, const int* __restrict__ degI,
                            float* __restrict__ invO, float* __restrict__ invI, int n) {
  int i = blockIdx.x * blockDim.x + threadIdx.x;
  if (i < n) {
    invO[i] = rsqrtf(fmaxf((float)degO[i], 1.0f));
    invI[i] = rsqrtf(fmaxf((float)degI[i], 1.0f));
  }
}

// ---------------- hs = feat * invOut[row], 128 cols, float4 granularity --------
__global__ void gcn_scale_rows128(const float4* __restrict__ f, const float* __restrict__ s,
                                  float4* __restrict__ out, int n4) {
  int i = blockIdx.x * blockDim.x + threadIdx.x;
  if (i < n4) {
    int row = i >> 5;              // 32 float4 per 128-col row
    float sc = s[row];
    float4 v = f[i];
    v.x *= sc; v.y *= sc; v.z *= sc; v.w *= sc;
    out[i] = v;
  }
}

// ---------------- edge scatter: one wave32 per edge, 128 f32 cols ----------------
__global__ void gcn_scatter128(const float* __restrict__ hs, const int* __restrict__ src,
                               const int* __restrict__ dst, float* __restrict__ agg, int ne) {
  int wid  = (blockIdx.x * blockDim.x + threadIdx.x) >> 5;
  int lane = threadIdx.x & 31;
  if (wid >= ne) return;
  int s = src[wid];
  int d = dst[wid];
  const float4 v = ((const float4*)(hs + (size_t)s * 128))[lane];   // coalesced b128 row read
  float* base = agg + (size_t)d * 128 + lane * 4;
  unsafeAtomicAdd(base + 0, v.x);
  unsafeAtomicAdd(base + 1, v.y);
  unsafeAtomicAdd(base + 2, v.z);
  unsafeAtomicAdd(base + 3, v.w);
}

// ---------------- WMMA f32 GEMM: out[rowsxND] = (A * invIn[row]) @ W + bias ------
// One wave computes a 16x16 f32 tile via V_WMMA_F32_16X16X4_F32, K = KD in steps of 4.
// RELU applies relu; SCALE_OUT multiplies stored value by invOut[row] (fuses next
// layer's source normalization into this GEMM's epilogue).
template <int KD, int ND, bool RELU, bool SCALE_OUT>
__global__ void gcn_gemm_wmma(const float* __restrict__ A, const float* __restrict__ invIn,
                              const float* __restrict__ invOut, const float* __restrict__ W,
                              const float* __restrict__ bias, float* __restrict__ out,
                              int nWaves) {
  constexpr int TN = ND / 16;
  int wid  = (blockIdx.x * blockDim.x + threadIdx.x) >> 5;
  int lane = threadIdx.x & 31;
  if (wid >= nWaves) return;                 // wave-uniform: EXEC stays all-ones
  int tileM = wid / TN;
  int tileN = wid - tileM * TN;

  int rlane = lane & 15;
  int khalf = lane >> 4;                     // 0: K=k0..k0+1, 1: K=k0+2..k0+3
  int rowA  = tileM * 16 + rlane;            // A fragment row for this lane
  int colB  = tileN * 16 + rlane;            // B/C/D column for this lane
  float sIn = invIn[rowA];

  const float* arow = A + (size_t)rowA * KD;
  v8f acc = {};
#pragma unroll
  for (int k0 = 0; k0 < KD; k0 += 4) {
    int k = k0 + khalf * 2;
    const float2 av = *(const float2*)(arow + k);        // 8B aligned (k even)
    v2f a; a.x = av.x * sIn; a.y = av.y * sIn;
    v2f b;
    b.x = W[(size_t)k * ND + colB];
    b.y = W[(size_t)(k + 1) * ND + colB];
    acc = __builtin_amdgcn_wmma_f32_16x16x4_f32(
        /*neg_a=*/false, a, /*neg_b=*/false, b,
        /*c_mod=*/(short)0, acc, /*reuse_a=*/false, /*reuse_b=*/false);
  }

  float bv = bias[colB];
#pragma unroll
  for (int i = 0; i < 8; ++i) {
    int row = tileM * 16 + khalf * 8 + i;    // C/D layout: VGPR i -> M = i + 8*khalf
    float v = acc[i] + bv;
    if (RELU) v = fmaxf(v, 0.0f);
    if (SCALE_OUT) v *= invOut[row];
    out[(size_t)row * ND + colB] = v;
  }
}

extern "C" void kernel_launch(void* const* d_in, const int* in_sizes, int n_in,
                              void* d_out, int out_size, void* d_ws, size_t ws_size,
                              hipStream_t stream) {
  const float* feat = (const float*)d_in[0];   // [N,128]
  const int*   src  = (const int*)  d_in[1];   // [E]
  const int*   dst  = (const int*)  d_in[2];   // [E]
  const float* W1   = (const float*)d_in[3];   // [128,128] row-major (K x N)
  const float* b1   = (const float*)d_in[4];   // [128]
  const float* W2   = (const float*)d_in[5];   // [128,64]
  const float* b2   = (const float*)d_in[6];   // [64]
  float* out = (float*)d_out;                  // [N,64]

  float* ws   = (float*)d_ws;
  int*   degO = (int*)ws;                       // [N] ints
  int*   degI = degO + NN;                      // [N] ints
  float* invO = ws + 2 * (size_t)NN;            // [N]
  float* invI = invO + NN;                      // [N]
  float* bufA = ws + 4 * (size_t)NN;            // [N,128]  hs1 -> hs2
  float* bufB = bufA + (size_t)NN * 128;        // [N,128]  agg1 -> agg2

  const int nFeat4 = NN * 32;                   // N*128/4 float4
  const int scatterBlocks = (NE * 32 + 255) / 256;

  // 1) degrees
  gcn_zero_f4<<<(2 * NN / 4 + 255) / 256, 256, 0, stream>>>((float4*)degO, 2 * NN / 4);
  gcn_degrees<<<(NE + 255) / 256, 256, 0, stream>>>(src, dst, degO, degI, NE);
  gcn_invsqrt<<<(NN + 255) / 256, 256, 0, stream>>>(degO, degI, invO, invI, NN);

  // 2) hs1 = feat * invO
  gcn_scale_rows128<<<(nFeat4 + 255) / 256, 256, 0, stream>>>(
      (const float4*)feat, invO, (float4*)bufA, nFeat4);

  // 3) agg1 = scatter-add(hs1)
  gcn_zero_f4<<<(nFeat4 + 255) / 256, 256, 0, stream>>>((float4*)bufB, nFeat4);
  gcn_scatter128<<<scatterBlocks, 256, 0, stream>>>(bufA, src, dst, bufB, NE);

  // 4) hs2 = relu((agg1*invI) @ W1 + b1) * invO    (layer-2 pre-scale fused)
  {
    int nWaves = (NN / 16) * (HID_D / 16);     // 6250 * 8
    gcn_gemm_wmma<IN_D, HID_D, true, true>
        <<<nWaves * 32 / 256, 256, 0, stream>>>(bufB, invI, invO, W1, b1, bufA, nWaves);
  }

  // 5) agg2 = scatter-add(hs2)   (reuse bufB)
  gcn_zero_f4<<<(nFeat4 + 255) / 256, 256, 0, stream>>>((float4*)bufB, nFeat4);
  gcn_scatter128<<<scatterBlocks, 256, 0, stream>>>(bufA, src, dst, bufB, NE);

  // 6) out = (agg2*invI) @ W2 + b2
  {
    int nWaves = (NN / 16) * (OUT_D / 16);     // 6250 * 4
    gcn_gemm_wmma<HID_D, OUT_D, false, false>
        <<<nWaves * 32 / 256, 256, 0, stream>>>(bufB, invI, invO, W2, b2, out, nWaves);
  }
}